// Network_75926431858958
// MI455X (gfx1250) — compile-verified
//
#include <hip/hip_runtime.h>

// Spiking-ensemble scan: per-pixel recurrence over T=100 steps.
// Pure streaming kernel: 2.0 GB total traffic -> ~86 us roofline at 23.3 TB/s.
// No matmul in this op => no WMMA; optimization target is the VMEM path:
// b128 non-temporal loads/stores + register-pipelined next-plane load.

typedef __attribute__((ext_vector_type(4))) float v4f;

#define T_STEPS 100

__constant__ constexpr float BETA          = 0.9f;
__constant__ constexpr float FREQ_BETA     = 0.95f;
__constant__ constexpr float ONE_M_FB      = 1.0f - 0.95f;   // 0.05
__constant__ constexpr float BASE_THRESH   = 1.0f;
__constant__ constexpr float TARGET_FREQ   = 0.1f;
__constant__ constexpr float THR_INC       = 0.05f;
__constant__ constexpr float INV_THR_DIV   = 1.0f / 1.05f;   // compile-time reciprocal

__global__ __launch_bounds__(256) void snn_scan_kernel(
    const float* __restrict__ x,   // (T, N) flattened
    float* __restrict__ out,       // spk | act | thr | freq, each (T, N)
    int N)                         // pixels per plane (H*W), multiple of 4
{
    const int base = (blockIdx.x * blockDim.x + threadIdx.x) * 4;
    if (base >= N) return;

    float* __restrict__ spk_o = out;
    float* __restrict__ act_o = out + (size_t)1 * T_STEPS * N;
    float* __restrict__ thr_o = out + (size_t)2 * T_STEPS * N;
    float* __restrict__ frq_o = out + (size_t)3 * T_STEPS * N;

    // Per-pixel state lives in VGPRs for the whole scan.
    float act[4], thr[4], frq[4];
#pragma unroll
    for (int k = 0; k < 4; ++k) {
        act[k] = 0.0f;
        thr[k] = BASE_THRESH;
        frq[k] = 0.0f;
    }

    // Software pipeline: keep next time-plane load in flight (register
    // double-buffer; no LDS staging needed for a use-once coalesced stream).
    v4f xv = __builtin_nontemporal_load(
        reinterpret_cast<const v4f*>(x + base));

    for (int t = 0; t < T_STEPS; ++t) {
        v4f xn = xv;
        if (t + 1 < T_STEPS) {
            xn = __builtin_nontemporal_load(
                reinterpret_cast<const v4f*>(x + (size_t)(t + 1) * N + base));
        }

        const float xin[4] = {xv.x, xv.y, xv.z, xv.w};
        float sv[4], av[4], tv[4], fv[4];

#pragma unroll
        for (int k = 0; k < 4; ++k) {
            // membrane EMA + input
            float a = fmaf(BETA, act[k], xin[k]);
            // spike against *old* threshold
            const bool spike = a > thr[k];
            const float s = spike ? 1.0f : 0.0f;
            // running-average spike frequency
            const float f = fmaf(FREQ_BETA, frq[k], ONE_M_FB * s);
            // auto gain control: the two sequential wheres have disjoint
            // predicates, so a ternary chain is exactly equivalent.
            float th = thr[k];
            th = (f > TARGET_FREQ) ? (th + THR_INC)
               : ((f < TARGET_FREQ) ? (th * INV_THR_DIV) : th);
            // zero reset on spike (after threshold update, per reference order)
            a = spike ? 0.0f : a;

            act[k] = a; thr[k] = th; frq[k] = f;
            sv[k] = s;  av[k] = a;  tv[k] = th; fv[k] = f;
        }

        const size_t off = (size_t)t * N + base;
        v4f svv = {sv[0], sv[1], sv[2], sv[3]};
        v4f avv = {av[0], av[1], av[2], av[3]};
        v4f tvv = {tv[0], tv[1], tv[2], tv[3]};
        v4f fvv = {fv[0], fv[1], fv[2], fv[3]};
        // Write-once histories: NT stores keep the 1.6 GB output stream from
        // polluting WGP$/L2.
        __builtin_nontemporal_store(svv, reinterpret_cast<v4f*>(spk_o + off));
        __builtin_nontemporal_store(avv, reinterpret_cast<v4f*>(act_o + off));
        __builtin_nontemporal_store(tvv, reinterpret_cast<v4f*>(thr_o + off));
        __builtin_nontemporal_store(fvv, reinterpret_cast<v4f*>(frq_o + off));

        xv = xn;
    }
}

extern "C" void kernel_launch(void* const* d_in, const int* in_sizes, int n_in,
                              void* d_out, int out_size, void* d_ws, size_t ws_size,
                              hipStream_t stream) {
    (void)n_in; (void)out_size; (void)d_ws; (void)ws_size;
    const float* x = (const float*)d_in[0];
    float* out = (float*)d_out;

    const int total = in_sizes[0];        // T * H * W
    const int N = total / T_STEPS;        // H * W  (= 1M, multiple of 4)

    const int threads = 256;
    const int vec_threads = N / 4;        // one thread per 4 pixels
    const int blocks = (vec_threads + threads - 1) / threads;

    snn_scan_kernel<<<blocks, threads, 0, stream>>>(x, out, N);
}